// BalSCL_3616362463447
// MI455X (gfx1250) — compile-verified
//
#include <hip/hip_runtime.h>
#include <math.h>

typedef float        v2f  __attribute__((ext_vector_type(2)));
typedef float        v8f  __attribute__((ext_vector_type(8)));
typedef unsigned int u32x4 __attribute__((ext_vector_type(4)));
typedef int          i32x4 __attribute__((ext_vector_type(4)));
typedef int          i32x8 __attribute__((ext_vector_type(8)));

#define BATCH      4096
#define TWOB       8192
#define FEATSTRIDE 512     // features row stride (2 views * 256)
#define DV         256     // per-view dim (K of the GEMM)
#define NCLS       100
#define INV_T      10.0f

#define ROWS_PER_BLK 128
#define COLSPLIT     8
#define COLS_PER_BLK 1024   // 8 column-tiles of 128
#define CT_WIDTH     128
#define KC           64
#define A_STRIDE     260    // 256 + 4 pad dwords  (260 % 64 == 4 -> conflict-free b64)
#define B_STRIDE     68     // 64 + 4 pad dwords   (68  % 64 == 4)

// workspace layout (floats)
#define WS_INV   0                       // [128]  1/(2*count[class])
#define WS_BLK   128                     // [64]   per-merge-block (num,den)
#define WS_PART  256                     // [8192*8*4] per-(row,split) m,s,p,cnt

// ---------------- TDM: 2D strided tile -> LDS with row padding ----------------
// tile_x elements (4B) per row, tile_y rows, global row stride = 512 floats.
// pad_code_i: LDS pad interval code (5 => every 64 dwords, 7 => every 256 dwords)
// pad_code_a: LDS pad amount code  (3 => 4 dwords)
__device__ __forceinline__ void tdm_load_2d(unsigned lds_off, const float* gptr,
                                            int tile_x, int tile_y,
                                            int pad_code_i, int pad_code_a) {
    unsigned long long ga = (unsigned long long)(size_t)gptr;
    u32x4 g0;
    g0.x = 1u;                                    // count=1, user descriptor
    g0.y = lds_off;                               // lds_addr (bytes)
    g0.z = (unsigned)ga;                          // global_addr[31:0]
    g0.w = (unsigned)((ga >> 32) & 0x01ffffffu)   // global_addr[56:32]
           | 0x80000000u;                         // type=2 ("image")
    i32x8 g1 = {0, 0, 0, 0, 0, 0, 0, 0};
    g1[0] = (2 << 16)                             // data_size = 4 bytes
          | (1 << 20)                             // pad_enable
          | (pad_code_i << 22)                    // pad_interval
          | (pad_code_a << 25);                   // pad_amount
    g1[1] = (FEATSTRIDE << 16);                   // tensor_dim0 lo16 = 512
    g1[2] = 0;                                    // tensor_dim0 hi, tensor_dim1 lo
    g1[3] = 0x10 | (tile_x << 16);                // tensor_dim1 hi (=> 1M rows), tile_dim0
    g1[4] = tile_y & 0xffff;                      // tile_dim1 (tile_dim2 = 0)
    g1[5] = FEATSTRIDE;                           // tensor_dim0_stride lo32 = 512
    i32x4 z4 = {0, 0, 0, 0};
#if __clang_major__ >= 23
    i32x8 z8 = {0, 0, 0, 0, 0, 0, 0, 0};
    __builtin_amdgcn_tensor_load_to_lds(g0, g1, z4, z4, z8, 0);
#else
    __builtin_amdgcn_tensor_load_to_lds(g0, g1, z4, z4, 0);
#endif
}

// ---------------- kernel 1: class counts -> inverse counts ----------------
__global__ __launch_bounds__(256) void bal_prep(const int* __restrict__ tgt,
                                                float* __restrict__ wsf) {
    __shared__ int cnt[NCLS];
    int t = threadIdx.x;
    if (t < NCLS) cnt[t] = 0;
    __syncthreads();
    for (int i = t; i < BATCH; i += 256) atomicAdd(&cnt[tgt[i]], 1);
    __syncthreads();
    if (t < 128) wsf[WS_INV + t] = (t < NCLS) ? 1.0f / (2.0f * (float)cnt[t]) : 0.0f;
}

// ------------- kernel 2: TDM-staged WMMA GEMM + online softmax partials -------------
__global__ __launch_bounds__(256) void bal_main(const float* __restrict__ F,
                                                const int* __restrict__ tgt,
                                                float* __restrict__ wsf) {
    __shared__ float As[ROWS_PER_BLK * A_STRIDE];      // As[row*260 + k], k=0..255
    __shared__ float Bs[2][CT_WIDTH * B_STRIDE];       // Bs[b][col*68 + k], k chunk of 64
    __shared__ float invcS[128];

    const int tid = threadIdx.x;
    const int r0  = blockIdx.x * ROWS_PER_BLK;
    const int c0  = blockIdx.y * COLS_PER_BLK;

    const unsigned asOff = (unsigned)(size_t)(void*)As;
    const unsigned bsOff[2] = { (unsigned)(size_t)(void*)Bs[0],
                                (unsigned)(size_t)(void*)Bs[1] };

    if (tid < 128) invcS[tid] = wsf[WS_INV + tid];

    const int w  = tid >> 5;          // wave 0..7 -> rows w*16..w*16+15
    const int l  = tid & 31;
    const int lh = l >> 4;            // lane half (M bias of 8, K bias of 2)
    const int ln = l & 15;

    // A tile via TDM: rows r0..r0+127 of feats (one view; r0 never straddles 4096)
    if (w == 0) {
        const float* ga = F + (size_t)(r0 & (BATCH - 1)) * FEATSTRIDE + ((r0 >> 12) << 8);
        tdm_load_2d(asOff, ga, DV, ROWS_PER_BLK, /*pad every 256 dw*/ 7, /*4 dw*/ 3);
    }

    int   iRow[8], cRow[8];
    float m[8], s[8], p[8], cm[8];
#pragma unroll
    for (int r = 0; r < 8; ++r) {
        iRow[r] = r0 + w * 16 + r + 8 * lh;
        cRow[r] = tgt[iRow[r] & (BATCH - 1)];
        m[r] = -INFINITY; s[r] = 0.0f; p[r] = 0.0f; cm[r] = 0.0f;
    }

    const int rowA = w * 16 + ln;     // A fragment: M = l%16
    const float* AsRow = &As[rowA * A_STRIDE + 2 * lh];

    for (int ct = 0; ct < COLS_PER_BLK / CT_WIDTH; ++ct) {
        const int j0 = c0 + ct * CT_WIDTH;
        const float* gb = F + (size_t)(j0 & (BATCH - 1)) * FEATSTRIDE + ((j0 >> 12) << 8);

        if (w == 0) tdm_load_2d(bsOff[0], gb, KC, CT_WIDTH, /*64 dw*/ 5, /*4 dw*/ 3);

        v8f acc[8] = {};

#pragma unroll
        for (int c = 0; c < 4; ++c) {              // K chunks of 64
            if (w == 0) {
                if (c < 3) {                       // prefetch next chunk, keep 1 in flight
                    tdm_load_2d(bsOff[(c + 1) & 1], gb + (c + 1) * KC, KC, CT_WIDTH, 5, 3);
                    __builtin_amdgcn_s_wait_tensorcnt(1);
                } else {                           // last chunk: drain
                    __builtin_amdgcn_s_wait_tensorcnt(0);
                }
            }
            __syncthreads();                       // chunk c resident in Bs[c&1]

            const float* BsBuf = &Bs[c & 1][2 * lh];
#pragma unroll
            for (int ks = 0; ks < KC; ks += 4) {
                v2f a = *(const v2f*)(AsRow + c * KC + ks);     // K = r + 2*(l/16)
#pragma unroll
                for (int nt = 0; nt < 8; ++nt) {
                    const int col = nt * 16 + ln;               // N = l%16
                    v2f b = *(const v2f*)(BsBuf + col * B_STRIDE + ks);
                    acc[nt] = __builtin_amdgcn_wmma_f32_16x16x4_f32(
                        false, a, false, b, (short)0, acc[nt], false, false);
                }
            }
            __syncthreads();                       // done reading Bs[c&1]
        }

        // per-lane column metadata for this 128-col tile
        int   jn[8], cj[8];
        float wj[8];
#pragma unroll
        for (int nt = 0; nt < 8; ++nt) {
            jn[nt] = j0 + nt * 16 + ln;
            cj[nt] = tgt[jn[nt] & (BATCH - 1)];
            wj[nt] = invcS[cj[nt]];
        }

        // online softmax update, 8 rows per lane (C layout: M = r + 8*lh)
#pragma unroll
        for (int r = 0; r < 8; ++r) {
            float v[8], tmax = -INFINITY;
#pragma unroll
            for (int nt = 0; nt < 8; ++nt) {
                v[nt] = acc[nt][r] * INV_T;
                tmax  = fmaxf(tmax, v[nt]);
            }
            for (int d = 1; d < 16; d <<= 1)
                tmax = fmaxf(tmax, __shfl_xor(tmax, d, 32));
            const float mnew = fmaxf(m[r], tmax);
            s[r] *= __expf(m[r] - mnew);           // exp(-inf)=0 on first tile
            m[r]  = mnew;
            const int i = iRow[r];
#pragma unroll
            for (int nt = 0; nt < 8; ++nt) {
                const bool self = (jn[nt] == i);
                const float e   = __expf(v[nt] - mnew);
                if (!self) s[r] += e * wj[nt];
                if (!self && (cj[nt] == cRow[r])) { p[r] += v[nt]; cm[r] += 1.0f; }
            }
        }
    }

    // reduce s/p/cnt across the 16 lanes of each row; write partials
#pragma unroll
    for (int r = 0; r < 8; ++r) {
        float ss = s[r], pp = p[r], cc = cm[r];
        for (int d = 1; d < 16; d <<= 1) {
            ss += __shfl_xor(ss, d, 32);
            pp += __shfl_xor(pp, d, 32);
            cc += __shfl_xor(cc, d, 32);
        }
        if (ln == 0) {
            float* dst = wsf + WS_PART +
                         ((size_t)iRow[r] * COLSPLIT + blockIdx.y) * 4;
            dst[0] = m[r]; dst[1] = ss; dst[2] = pp; dst[3] = cc;
        }
    }
}

// ------------- kernel 3: merge column-splits per row, deterministic block sums -------------
__global__ __launch_bounds__(256) void bal_merge(float* __restrict__ wsf) {
    const int i = blockIdx.x * 256 + threadIdx.x;   // row, < 8192
    const float* part = wsf + WS_PART + (size_t)i * COLSPLIT * 4;
    float mg = -INFINITY;
#pragma unroll
    for (int k = 0; k < COLSPLIT; ++k) mg = fmaxf(mg, part[k * 4]);
    float S = 0.0f, P = 0.0f, C = 0.0f;
#pragma unroll
    for (int k = 0; k < COLSPLIT; ++k) {
        S += part[k * 4 + 1] * __expf(part[k * 4] - mg);
        P += part[k * 4 + 2];
        C += part[k * 4 + 3];
    }
    float num = P - C * (mg + __logf(S));
    float den = C;
    for (int d = 1; d < 32; d <<= 1) {
        num += __shfl_xor(num, d, 32);
        den += __shfl_xor(den, d, 32);
    }
    __shared__ float red[16];
    const int w = threadIdx.x >> 5;
    if ((threadIdx.x & 31) == 0) { red[w] = num; red[8 + w] = den; }
    __syncthreads();
    if (threadIdx.x == 0) {
        float n = 0.0f, dd = 0.0f;
        for (int k = 0; k < 8; ++k) { n += red[k]; dd += red[8 + k]; }
        wsf[WS_BLK + blockIdx.x * 2]     = n;
        wsf[WS_BLK + blockIdx.x * 2 + 1] = dd;
    }
}

// ------------- kernel 4: finalize scalar -------------
__global__ void bal_final(const float* __restrict__ wsf, float* __restrict__ out) {
    float n = 0.0f, d = 0.0f;
    for (int b = 0; b < TWOB / 256; ++b) {
        n += wsf[WS_BLK + b * 2];
        d += wsf[WS_BLK + b * 2 + 1];
    }
    out[0] = -n / d;
}

extern "C" void kernel_launch(void* const* d_in, const int* in_sizes, int n_in,
                              void* d_out, int out_size, void* d_ws, size_t ws_size,
                              hipStream_t stream) {
    const float* features = (const float*)d_in[0];
    const int*   targets  = (const int*)d_in[1];
    float*       wsf      = (float*)d_ws;
    float*       out      = (float*)d_out;

    bal_prep<<<1, 256, 0, stream>>>(targets, wsf);
    dim3 grid(TWOB / ROWS_PER_BLK, COLSPLIT);   // 64 x 8 = 512 blocks
    bal_main<<<grid, 256, 0, stream>>>(features, targets, wsf);
    bal_merge<<<TWOB / 256, 256, 0, stream>>>(wsf);
    bal_final<<<1, 1, 0, stream>>>(wsf, out);
}